// MLPWithEdge_70892730187950
// MI455X (gfx1250) — compile-verified
//
#include <hip/hip_runtime.h>
#include <cstddef>

#define N_NODES_C  100000
#define N_EDGES_C  3200000
#define NODE_DIM_C 128
#define EDGE_DIM_C 16
#define CAT_DIM_C  144
#define OUT_DIM_C  128
#define BN_EPS_C   1e-5f

typedef float v2f __attribute__((ext_vector_type(2)));
typedef float v8f __attribute__((ext_vector_type(8)));

// ---------------------------------------------------------------- fill zeros
__global__ void fill_zero_kernel(float* __restrict__ p, int n) {
  int i = blockIdx.x * blockDim.x + threadIdx.x;
  if (i < n) p[i] = 0.0f;
}

// ---------------------------------------------------------------- edge scatter (sums + counts)
__global__ void scatter_kernel(const float* __restrict__ edge_attr,
                               const long long* __restrict__ seg,   // edge_index row 0 (int64)
                               float* __restrict__ sums,
                               float* __restrict__ cnt) {
  int e = blockIdx.x * blockDim.x + threadIdx.x;
  if (e >= N_EDGES_C) return;
  long long s = seg[e];
  const float4* v = reinterpret_cast<const float4*>(edge_attr + (size_t)e * EDGE_DIM_C);
  float* dst = sums + (size_t)s * EDGE_DIM_C;
#pragma unroll
  for (int q = 0; q < 4; ++q) {
    float4 t = v[q];
    atomicAdd(dst + 4 * q + 0, t.x);
    atomicAdd(dst + 4 * q + 1, t.y);
    atomicAdd(dst + 4 * q + 2, t.z);
    atomicAdd(dst + 4 * q + 3, t.w);
  }
  atomicAdd(cnt + s, 1.0f);
}

// ---------------------------------------------------------------- h0 = concat(x, sums/max(cnt,1))
__global__ void build_h0_kernel(const float* __restrict__ x,
                                const float* __restrict__ sums,
                                const float* __restrict__ cnt,
                                float* __restrict__ h0) {
  int i = blockIdx.x * blockDim.x + threadIdx.x;
  int n = i / CAT_DIM_C;
  int d = i - n * CAT_DIM_C;
  if (n >= N_NODES_C) return;
  float v;
  if (d < NODE_DIM_C) {
    v = x[(size_t)n * NODE_DIM_C + d];
  } else {
    v = sums[(size_t)n * EDGE_DIM_C + (d - NODE_DIM_C)] / fmaxf(cnt[n], 1.0f);
  }
  h0[(size_t)n * CAT_DIM_C + d] = v;
}

// ---------------------------------------------------------------- WMMA fp32 GEMM (BN pre-folded):
// out[N,128] = act( in[N,K] @ W[128,K]^T + b ), act = relu if do_relu
// One block = 128 threads (4 waves). Block covers 64 rows x 64 output cols.
// Each wave: one 16-row block, 4 column tiles of 16 (4x v8f accumulators).
__global__ void gemm_wmma_kernel(const float* __restrict__ in, int K,
                                 const float* __restrict__ W,
                                 const float* __restrict__ bias,
                                 float* __restrict__ out, int do_relu) {
  // LDS weight tile: 64 output rows of W, padded stride K+4 (bank-conflict free
  // for K=128 and K=144 with 8-byte lane reads).
  __shared__ float ldsW[64 * (CAT_DIM_C + 4)];
  const int Kp = K + 4;

  const int tid = threadIdx.x;           // 0..127
  const int col_half = blockIdx.x & 1;   // which 64 output channels
  const int row_super = blockIdx.x >> 1; // which 64-row stripe
  const int col_base = col_half * 64;

  // cooperative stage of W[col_base .. col_base+63][0..K) into LDS
  for (int i = tid; i < 64 * K; i += 128) {
    int r = i / K;
    int c = i - r * K;
    ldsW[r * Kp + c] = W[(size_t)(col_base + r) * K + c];
  }
  __syncthreads();

  const int wave = tid >> 5;   // 0..3
  const int lane = tid & 31;
  const int lrow = lane & 15;
  const int khi = (lane >> 4) * 2;  // 0 for lanes 0-15, 2 for lanes 16-31
  const int row_blk = row_super * 64 + wave * 16;

  int arow = row_blk + lrow;
  if (arow >= N_NODES_C) arow = N_NODES_C - 1;  // clamp; stores are guarded
  const float* aptr = in + (size_t)arow * K + khi;

  v8f acc0 = {}, acc1 = {}, acc2 = {}, acc3 = {};

  for (int k0 = 0; k0 < K; k0 += 4) {
    v2f a = *reinterpret_cast<const v2f*>(aptr + k0);
    const float* wbase = &ldsW[(size_t)lrow * Kp + k0 + khi];
    v2f b0 = *reinterpret_cast<const v2f*>(wbase + 0 * 16 * Kp);
    v2f b1 = *reinterpret_cast<const v2f*>(wbase + 1 * 16 * Kp);
    v2f b2 = *reinterpret_cast<const v2f*>(wbase + 2 * 16 * Kp);
    v2f b3 = *reinterpret_cast<const v2f*>(wbase + 3 * 16 * Kp);
    acc0 = __builtin_amdgcn_wmma_f32_16x16x4_f32(false, a, false, b0, (short)0, acc0, false, false);
    acc1 = __builtin_amdgcn_wmma_f32_16x16x4_f32(false, a, false, b1, (short)0, acc1, false, false);
    acc2 = __builtin_amdgcn_wmma_f32_16x16x4_f32(false, a, false, b2, (short)0, acc2, false, false);
    acc3 = __builtin_amdgcn_wmma_f32_16x16x4_f32(false, a, false, b3, (short)0, acc3, false, false);
  }

  // epilogue: D layout — VGPR j: lanes 0-15 -> M=j, lanes 16-31 -> M=j+8; N=lane&15
  v8f accs[4] = {acc0, acc1, acc2, acc3};
  const int mofs = (lane >> 4) * 8;
#pragma unroll
  for (int c = 0; c < 4; ++c) {
    int col = col_base + c * 16 + lrow;
    float bb = bias[col];
#pragma unroll
    for (int j = 0; j < 8; ++j) {
      int row = row_blk + mofs + j;
      if (row < N_NODES_C) {
        float v = accs[c][j] + bb;
        if (do_relu) v = fmaxf(v, 0.0f);
        out[(size_t)row * OUT_DIM_C + col] = v;
      }
    }
  }
}

// ---------------------------------------------------------------- BN: per-channel sum / sumsq
__global__ void bn_reduce_kernel(const float* __restrict__ y,
                                 float* __restrict__ ssum, float* __restrict__ ssq) {
  int tid = threadIdx.x;          // 256 threads: 2 sub-rows x 128 channels
  int ch = tid & 127;
  int half = tid >> 7;
  float s = 0.0f, q = 0.0f;
  for (int n = blockIdx.x * 2 + half; n < N_NODES_C; n += gridDim.x * 2) {
    float v = y[(size_t)n * OUT_DIM_C + ch];
    s += v;
    q += v * v;
  }
  atomicAdd(&ssum[ch], s);
  atomicAdd(&ssq[ch], q);
}

// ---------------------------------------------------------------- BN: fold into scale/shift
__global__ void bn_stats_kernel(const float* __restrict__ ssum, const float* __restrict__ ssq,
                                const float* __restrict__ gamma, const float* __restrict__ beta,
                                float* __restrict__ scale, float* __restrict__ shift) {
  int ch = threadIdx.x;
  if (ch < OUT_DIM_C) {
    const float invN = 1.0f / (float)N_NODES_C;
    float mean = ssum[ch] * invN;
    float var = ssq[ch] * invN - mean * mean;
    float sc = gamma[ch] * rsqrtf(var + BN_EPS_C);
    scale[ch] = sc;
    shift[ch] = beta[ch] - mean * sc;
  }
}

// ---------------------------------------------------------------- fold BN(scale,shift) into W,b:
// (in*s + t) @ W^T + b  ==  in @ (W .* s)^T + (b + W @ t)
// One block per output row o (128 blocks x 128 threads, K==128).
__global__ void bn_fold_kernel(const float* __restrict__ W, const float* __restrict__ b,
                               const float* __restrict__ scale, const float* __restrict__ shift,
                               float* __restrict__ Wf, float* __restrict__ bf) {
  __shared__ float red[128];
  int o = blockIdx.x;
  int k = threadIdx.x;
  float w = W[(size_t)o * OUT_DIM_C + k];
  Wf[(size_t)o * OUT_DIM_C + k] = w * scale[k];
  red[k] = w * shift[k];
  __syncthreads();
#pragma unroll
  for (int off = 64; off > 0; off >>= 1) {
    if (k < off) red[k] += red[k + off];
    __syncthreads();
  }
  if (k == 0) bf[o] = b[o] + red[0];
}

// ================================================================ launch
extern "C" void kernel_launch(void* const* d_in, const int* in_sizes, int n_in,
                              void* d_out, int out_size, void* d_ws, size_t ws_size,
                              hipStream_t stream) {
  (void)in_sizes; (void)n_in; (void)out_size; (void)ws_size;

  const float*     x     = (const float*)d_in[0];
  const float*     ea    = (const float*)d_in[1];
  const long long* eidx  = (const long long*)d_in[2];   // int64 [2, E]; row 0 = seg
  const float*     W0    = (const float*)d_in[3];
  const float*     b0    = (const float*)d_in[4];
  const float*     W1    = (const float*)d_in[5];
  const float*     b1    = (const float*)d_in[6];
  const float*     W2    = (const float*)d_in[7];
  const float*     b2    = (const float*)d_in[8];
  const float*     Wout  = (const float*)d_in[9];
  const float*     bout  = (const float*)d_in[10];
  const float*     gamma = (const float*)d_in[11];
  const float*     beta  = (const float*)d_in[12];
  float* out = (float*)d_out;
  float* ws  = (float*)d_ws;

  // workspace layout (floats)
  float* sums = ws;                    //  1,600,000
  float* cnt  = ws + 1600000;          //    100,000
  float* h0   = ws + 1700000;          // 14,400,000 (aliased by y1 later)
  float* y0   = ws + 16100000;         // 12,800,000
  float* y1   = h0;                    // 12,800,000 (h0 is dead by then)
  float* ssum = ws + 28900000;         // 128
  float* ssq  = ssum + 128;            // 128
  float* bnsc = ssq + 128;             // 128
  float* bnsh = bnsc + 128;            // 128
  float* Wf   = bnsh + 128;            // 16,384
  float* bf   = Wf + 16384;            // 128

  const int rb = (N_NODES_C + 63) / 64;   // 64-row stripes
  const dim3 gemm_grid(rb * 2);           // x2 column halves
  const dim3 gemm_block(128);

  // 1) scatter-mean accumulators
  fill_zero_kernel<<<(1700000 + 255) / 256, 256, 0, stream>>>(sums, 1700000);
  scatter_kernel<<<(N_EDGES_C + 255) / 256, 256, 0, stream>>>(ea, eidx, sums, cnt);
  build_h0_kernel<<<(N_NODES_C * CAT_DIM_C + 255) / 256, 256, 0, stream>>>(x, sums, cnt, h0);

  // 2) layer 0: h0[144] -> y0 (relu), no input BN
  gemm_wmma_kernel<<<gemm_grid, gemm_block, 0, stream>>>(h0, CAT_DIM_C, W0, b0, y0, 1);
  fill_zero_kernel<<<1, 256, 0, stream>>>(ssum, 256);
  bn_reduce_kernel<<<512, 256, 0, stream>>>(y0, ssum, ssq);
  bn_stats_kernel<<<1, 128, 0, stream>>>(ssum, ssq, gamma, beta, bnsc, bnsh);

  // 3) layer 1: BN(y0)[128] -> y1 (relu); BN folded into W1,b1
  bn_fold_kernel<<<128, 128, 0, stream>>>(W1, b1, bnsc, bnsh, Wf, bf);
  gemm_wmma_kernel<<<gemm_grid, gemm_block, 0, stream>>>(y0, OUT_DIM_C, Wf, bf, y1, 1);
  fill_zero_kernel<<<1, 256, 0, stream>>>(ssum, 256);
  bn_reduce_kernel<<<512, 256, 0, stream>>>(y1, ssum, ssq);
  bn_stats_kernel<<<1, 128, 0, stream>>>(ssum, ssq, gamma, beta, bnsc, bnsh);

  // 4) layer 2: BN(y1)[128] -> y0 (relu); BN folded into W2,b2
  bn_fold_kernel<<<128, 128, 0, stream>>>(W2, b2, bnsc, bnsh, Wf, bf);
  gemm_wmma_kernel<<<gemm_grid, gemm_block, 0, stream>>>(y1, OUT_DIM_C, Wf, bf, y0, 1);
  fill_zero_kernel<<<1, 256, 0, stream>>>(ssum, 256);
  bn_reduce_kernel<<<512, 256, 0, stream>>>(y0, ssum, ssq);
  bn_stats_kernel<<<1, 128, 0, stream>>>(ssum, ssq, gamma, beta, bnsc, bnsh);

  // 5) output layer: BN(y0)[128] -> out (no relu); BN folded into Wout,bout
  bn_fold_kernel<<<128, 128, 0, stream>>>(Wout, bout, bnsc, bnsh, Wf, bf);
  gemm_wmma_kernel<<<gemm_grid, gemm_block, 0, stream>>>(y0, OUT_DIM_C, Wf, bf, out, 0);
}